// GCNBlock_MIX_20744692039829
// MI455X (gfx1250) — compile-verified
//
#include <hip/hip_runtime.h>

// Problem constants (from the reference harness)
#define NNODES 50000
#define NEDGES 800000
#define NGRAPH 64
#define D      96      // feature dim per conv
#define D2     192     // concat dim
#define KC     96      // K-chunk staged in LDS per pass
#define WSTRIDE 104    // LDS row stride (pad: 2*104 % 64banks = 16 -> lanes16-31 disjoint banks)

typedef float v2f __attribute__((ext_vector_type(2)));
typedef float v8f __attribute__((ext_vector_type(8)));

// ---------------- utility kernels ----------------

__global__ void k_zero(float* __restrict__ p, int n) {
    int i = blockIdx.x * blockDim.x + threadIdx.x;
    if (i < n) p[i] = 0.0f;
}

// deg[dst] += w  (edge_index flat as [2,E]: src = ei[0..E), dst = ei[E..2E))
__global__ void k_deg(const int* __restrict__ ei, const float* __restrict__ w,
                      float* __restrict__ deg, int e) {
    int i = blockIdx.x * blockDim.x + threadIdx.x;
    if (i < e) atomicAdd(&deg[ei[e + i]], w[i]);
}

// in-place deg -> rsqrt(deg + 1)  (self loop weight 1; deg+1 > 0 always)
__global__ void k_dinv(float* __restrict__ d, int n) {
    int i = blockIdx.x * blockDim.x + threadIdx.x;
    if (i < n) d[i] = rsqrtf(d[i] + 1.0f);
}

// norm[e] = dinv[src] * w[e] * dinv[dst]
__global__ void k_norm(const int* __restrict__ ei, const float* __restrict__ w,
                       const float* __restrict__ dinv, float* __restrict__ norm, int e) {
    int i = blockIdx.x * blockDim.x + threadIdx.x;
    if (i < e) {
        int s = ei[i], d = ei[e + i];
        norm[i] = dinv[s] * w[i] * dinv[d];
    }
}

__global__ void k_count(const int* __restrict__ batch, float* __restrict__ cnt, int n) {
    int i = blockIdx.x * blockDim.x + threadIdx.x;
    if (i < n) atomicAdd(&cnt[batch[i]], 1.0f);
}

// ---------------- WMMA fp32 GEMM (row-strip version) ----------------
// C[M x 96] = act(A[M x K] @ W[K x 96] (+bias)),  K in {96, 192}, M % 16 == 0.
// One wave computes a full 16x96 strip: 6 accumulators, A fragment reused 6x.
// W is staged chunk-by-chunk (96 x 96) in padded LDS shared by all 8 waves.
__global__ __launch_bounds__(256)
void k_wmma_gemm(const float* __restrict__ A, const float* __restrict__ Wg,
                 const float* __restrict__ bias, float* __restrict__ C,
                 int M, int K, int relu) {
    __shared__ float Wl[KC * WSTRIDE];               // ~40 KB

    const int lane   = threadIdx.x & 31;
    const int wave   = threadIdx.x >> 5;
    const int strip  = blockIdx.x * (blockDim.x >> 5) + wave;   // wave-uniform
    const int nstrip = M >> 4;

    const int half = lane >> 4;                      // 0: rows K+0/K+1 ; 1: rows K+2/K+3
    const int l16  = lane & 15;
    const float* __restrict__ arp = A + (size_t)(strip * 16 + l16) * K;

    v8f acc[6];
    #pragma unroll
    for (int t = 0; t < 6; ++t) acc[t] = (v8f){};

    for (int kc = 0; kc < K; kc += KC) {
        // cooperative fill of the 96x96 W chunk (float4, 16B-aligned: WSTRIDE*4 % 16 == 0)
        for (int i = threadIdx.x; i < KC * (D / 4); i += blockDim.x) {
            int r = i / (D / 4), c4 = (i - r * (D / 4)) * 4;
            *(float4*)(Wl + r * WSTRIDE + c4) = *(const float4*)(Wg + (size_t)(kc + r) * D + c4);
        }
        __syncthreads();

        if (strip < nstrip) {                        // wave-uniform guard: EXEC all-1s inside
            if (kc + KC < K) __builtin_prefetch(arp + kc + KC, 0, 1);  // global_prefetch_b8
            #pragma unroll 4
            for (int k0 = 0; k0 < KC; k0 += 4) {
                const int kk = k0 + half * 2;
                const float2 av = *(const float2*)(arp + kc + kk);   // 8B-aligned
                v2f a; a[0] = av.x; a[1] = av.y;
                #pragma unroll
                for (int t = 0; t < 6; ++t) {
                    v2f b;
                    b[0] = Wl[kk * WSTRIDE + t * 16 + l16];
                    b[1] = Wl[(kk + 1) * WSTRIDE + t * 16 + l16];
                    // 8 args: (neg_a, A, neg_b, B, c_mod, C, reuse_a, reuse_b)
                    acc[t] = __builtin_amdgcn_wmma_f32_16x16x4_f32(
                        false, a, false, b, (short)0, acc[t], false, false);
                }
            }
        }
        __syncthreads();                             // protect Wl before next fill
    }

    if (strip >= nstrip) return;

    // C/D layout: VGPR v -> row M = v + (lane<16 ? 0 : 8), col N = lane%16
    #pragma unroll
    for (int t = 0; t < 6; ++t) {
        const int col = t * 16 + l16;
        const float bv = bias ? bias[col] : 0.0f;
        #pragma unroll
        for (int v = 0; v < 8; ++v) {
            int row = strip * 16 + half * 8 + v;
            float x = acc[t][v] + bv;
            if (relu) x = fmaxf(x, 0.0f);
            C[(size_t)row * D + col] = x;
        }
    }
}

// ---------------- concat-buffer init: self-loop contribution + bias ----------------
__global__ void k_init_hcat(const float* __restrict__ xwo, const float* __restrict__ xwn,
                            const float* __restrict__ dinvo, const float* __restrict__ dinvn,
                            const float* __restrict__ bo, const float* __restrict__ bn,
                            float* __restrict__ hcat, int n) {
    int idx = blockIdx.x * blockDim.x + threadIdx.x;
    if (idx >= n * D) return;
    int i = idx / D, j = idx - i * D;
    float ao = dinvo[i]; ao *= ao;
    float an = dinvn[i]; an *= an;
    hcat[(size_t)i * D2 + j]     = ao * xwo[idx] + bo[j];
    hcat[(size_t)i * D2 + D + j] = an * xwn[idx] + bn[j];
}

// ---------------- edge scatter (wave-per-edge) ----------------
// One wave handles one edge: src/dst/norm are wave-uniform (scalarizable loads);
// each lane processes 3 consecutive floats (96 = 32*3) with f32 atomics into L2.
__global__ __launch_bounds__(256)
void k_scatter(const int* __restrict__ ei, const float* __restrict__ norm,
               const float* __restrict__ xw, float* __restrict__ hcat,
               int e, int coloff) {
    const int lane = threadIdx.x & 31;
    const int eidx = blockIdx.x * (blockDim.x >> 5) + (threadIdx.x >> 5);
    if (eidx >= e) return;
    const int s  = ei[eidx];
    const int d  = ei[e + eidx];
    const float nm = norm[eidx];
    const int f = lane * 3;
    const float* __restrict__ srcp = xw + (size_t)s * D + f;
    const float x0 = srcp[0], x1 = srcp[1], x2 = srcp[2];
    float* dst = hcat + (size_t)d * D2 + coloff + f;
    atomicAdd(dst + 0, nm * x0);
    atomicAdd(dst + 1, nm * x1);
    atomicAdd(dst + 2, nm * x2);
}

// ---------------- mean pool (sum via atomics, then divide) ----------------
__global__ void k_pool(const int* __restrict__ batch, const float* __restrict__ h,
                       float* __restrict__ out, int n) {
    int idx = blockIdx.x * blockDim.x + threadIdx.x;
    if (idx >= n * D) return;
    int i = idx / D, j = idx - i * D;
    atomicAdd(&out[(size_t)batch[i] * D + j], h[idx]);
}

__global__ void k_pooldiv(float* __restrict__ out, const float* __restrict__ cnt) {
    int idx = blockIdx.x * blockDim.x + threadIdx.x;
    if (idx >= NGRAPH * D) return;
    out[idx] /= fmaxf(cnt[idx / D], 1.0f);
}

// ---------------- host orchestration ----------------

static inline int cdiv(long long a, long long b) { return (int)((a + b - 1) / b); }

extern "C" void kernel_launch(void* const* d_in, const int* in_sizes, int n_in,
                              void* d_out, int out_size, void* d_ws, size_t ws_size,
                              hipStream_t stream) {
    (void)in_sizes; (void)n_in; (void)out_size; (void)ws_size;

    const float* x     = (const float*)d_in[0];
    const int*   ei    = (const int*)  d_in[1];   // [2,E] flat
    const float* ew    = (const float*)d_in[2];
    const int*   batch = (const int*)  d_in[3];
    const int*   ein   = (const int*)  d_in[4];
    const float* ewn   = (const float*)d_in[5];
    // d_in[6] (batch_neighbor) unused by the reference output path.

    const float* Wo[2] = { (const float*)d_in[7],  (const float*)d_in[13] };
    const float* bo[2] = { (const float*)d_in[8],  (const float*)d_in[14] };
    const float* Wn[2] = { (const float*)d_in[9],  (const float*)d_in[15] };
    const float* bn[2] = { (const float*)d_in[10], (const float*)d_in[16] };
    const float* Wf[2] = { (const float*)d_in[11], (const float*)d_in[17] };
    const float* bf[2] = { (const float*)d_in[12], (const float*)d_in[18] };

    float* out = (float*)d_out;

    // workspace carve-up (floats)
    float* ws = (float*)d_ws;
    size_t o = 0;
    float* dinv_o = ws + o; o += NNODES;
    float* dinv_n = ws + o; o += NNODES;
    float* norm_o = ws + o; o += NEDGES;
    float* norm_n = ws + o; o += NEDGES;
    float* cnt    = ws + o; o += NGRAPH;
    float* H      = ws + o; o += (size_t)NNODES * D;   // layer output / next input
    float* XWo    = ws + o; o += (size_t)NNODES * D;
    float* XWn    = ws + o; o += (size_t)NNODES * D;
    float* Hcat   = ws + o; o += (size_t)NNODES * D2;

    const int B = 256;

    // ---- per-call precompute shared by both layers ----
    k_zero<<<cdiv(NNODES, B), B, 0, stream>>>(dinv_o, NNODES);
    k_zero<<<cdiv(NNODES, B), B, 0, stream>>>(dinv_n, NNODES);
    k_zero<<<cdiv(NGRAPH, B), B, 0, stream>>>(cnt, NGRAPH);
    k_zero<<<cdiv(2 * NGRAPH * D, B), B, 0, stream>>>(out, 2 * NGRAPH * D);

    k_deg<<<cdiv(NEDGES, B), B, 0, stream>>>(ei,  ew,  dinv_o, NEDGES);
    k_deg<<<cdiv(NEDGES, B), B, 0, stream>>>(ein, ewn, dinv_n, NEDGES);
    k_dinv<<<cdiv(NNODES, B), B, 0, stream>>>(dinv_o, NNODES);
    k_dinv<<<cdiv(NNODES, B), B, 0, stream>>>(dinv_n, NNODES);
    k_norm<<<cdiv(NEDGES, B), B, 0, stream>>>(ei,  ew,  dinv_o, norm_o, NEDGES);
    k_norm<<<cdiv(NEDGES, B), B, 0, stream>>>(ein, ewn, dinv_n, norm_n, NEDGES);
    k_count<<<cdiv(NNODES, B), B, 0, stream>>>(batch, cnt, NNODES);

    // GEMM launch geometry: 3125 row-strips, 8 waves (strips) per block
    const int nstrip = NNODES / 16;
    const int gemmGB = cdiv(nstrip, 8);
    // Scatter: one wave per edge, 8 waves per block
    const int scatGB = cdiv(NEDGES, 8);

    for (int l = 0; l < 2; ++l) {
        const float* in = (l == 0) ? x : H;

        // XWo = in @ Wo ; XWn = in @ Wn   (fp32 WMMA, K=96)
        k_wmma_gemm<<<gemmGB, 256, 0, stream>>>(in, Wo[l], nullptr, XWo, NNODES, D, 0);
        k_wmma_gemm<<<gemmGB, 256, 0, stream>>>(in, Wn[l], nullptr, XWn, NNODES, D, 0);

        // Hcat = self-loop terms + biases, then scatter edge contributions
        k_init_hcat<<<cdiv((long long)NNODES * D, B), B, 0, stream>>>(
            XWo, XWn, dinv_o, dinv_n, bo[l], bn[l], Hcat, NNODES);
        k_scatter<<<scatGB, 256, 0, stream>>>(ei,  norm_o, XWo, Hcat, NEDGES, 0);
        k_scatter<<<scatGB, 256, 0, stream>>>(ein, norm_n, XWn, Hcat, NEDGES, D);

        // H = relu(Hcat @ Wf + bf)   (fp32 WMMA, K=192, two LDS chunks)
        k_wmma_gemm<<<gemmGB, 256, 0, stream>>>(Hcat, Wf[l], bf[l], H, NNODES, D2, 1);

        // mean pool into d_out[l]
        k_pool<<<cdiv((long long)NNODES * D, B), B, 0, stream>>>(
            batch, H, out + (size_t)l * NGRAPH * D, NNODES);
        k_pooldiv<<<cdiv(NGRAPH * D, B), B, 0, stream>>>(out + (size_t)l * NGRAPH * D, cnt);
    }
}